// HybridPixelGraphNca_14207751815643
// MI455X (gfx1250) — compile-verified
//
#include <hip/hip_runtime.h>

typedef __attribute__((ext_vector_type(2))) float v2f;
typedef __attribute__((ext_vector_type(8))) float v8f;

#define NPIX 65536
#define DD 12

// ---------------------------------------------------------------------------
// Pack B matrices into per-chunk WMMA lane layout.
// 36 chunks * 64 floats. For chunk q, lane L (0..31), slot v (0/1):
//   pack[q*64 + v*32 + L] = B(k0(q) + 2*(L>>4) + v, n = L&15)
// q 0..29 : GEMM1, K=120 : k<108 -> Wh im2col col, k>=108 -> Wmp^T
// q 30..32: GEMM2, K=12  : n<12 -> Wr1[:, :12]^T, n==12 -> Wa
// q 33..35: GEMM3, K=12  : n<3  -> Wr2^T
// ---------------------------------------------------------------------------
__global__ void build_bpack(const float* __restrict__ Wh,
                            const float* __restrict__ Wmp,
                            const float* __restrict__ Wr1,
                            const float* __restrict__ Wa,
                            const float* __restrict__ Wr2,
                            float* __restrict__ bpack) {
  for (int e = threadIdx.x; e < 36 * 64; e += blockDim.x) {
    int q = e >> 6;
    int s = e & 63;
    int L = s & 31;
    int n = L & 15;
    int kk = 2 * ((L >> 4) & 1) + (s >> 5);
    float v = 0.f;
    if (q < 30) {
      int k = 4 * q + kk;
      if (k < 108) {
        int t = k / 12, c = k - 12 * t;     // tap, in-channel
        int ky = t / 3, kx = t - 3 * ky;
        if (n < DD) v = Wh[((n * DD + c) * 3 + ky) * 3 + kx];
      } else {
        int c = k - 108;
        if (n < DD) v = Wmp[n * DD + c];
      }
    } else if (q < 33) {
      int k = 4 * (q - 30) + kk;            // 0..11
      if (n < DD) v = Wr1[n * 13 + k];
      else if (n == 12) v = Wa[k];
    } else {
      int k = 4 * (q - 33) + kk;            // 0..11
      if (n < 3) v = Wr2[n * DD + k];
    }
    bpack[e] = v;
  }
}

// ---------------------------------------------------------------------------
// One NCA step. Block = 256 threads = 8 waves; tile = 32(x) x 4(y) pixels,
// wave w handles row (w>>1), x-segment 16*(w&1). Each wave does three
// M=16-pixel x N=16 GEMMs with v_wmma_f32_16x16x4_f32.
// LDS hidden layout: sHid[c*204 + yy*34 + xx], halo tile 6 x 34 per channel.
// Half-1 lanes of the A operand always sit exactly 2 channels (+408 floats)
// above half-0 lanes, so A addresses are (baseH + compile-time-const).
// ---------------------------------------------------------------------------
__global__ __launch_bounds__(256)
void nca_step(const float* __restrict__ in, int inC, int inOff,
              float* __restrict__ out, int outFull,
              const float* __restrict__ bpack,
              const float* __restrict__ bh, const float* __restrict__ bmp,
              const float* __restrict__ Wr1, const float* __restrict__ ba,
              const float* __restrict__ br1, const float* __restrict__ br2) {
  __shared__ float sHid[12 * 6 * 34];   // [c][yy 0..5][xx 0..33] halo tile
  __shared__ float sT[8 * 256];         // per-wave 16x16 staging

  const int tid  = threadIdx.x;
  const int lane = tid & 31;
  const int wv   = tid >> 5;
  const int half = lane >> 4;
  const int nn   = lane & 15;

  const int x0 = blockIdx.x * 32;
  const int y0 = blockIdx.y * 4;
  const int b  = blockIdx.z;

  const size_t inBase = (size_t)(b * inC + inOff) * NPIX;

  // cooperative halo load (zero-padded): 204 threads each load 12 channels
  if (tid < 204) {
    int yy0 = tid / 34;
    int xx0 = tid - 34 * yy0;
    int gy = y0 - 1 + yy0;
    int gx = x0 - 1 + xx0;
    bool ok = ((unsigned)gy < 256u) && ((unsigned)gx < 256u);
    size_t gidx = inBase + (size_t)(gy * 256 + gx);
    #pragma unroll
    for (int c = 0; c < 12; ++c) {
      float v = 0.f;
      if (ok) v = in[gidx + (size_t)c * NPIX];
      sHid[c * 204 + tid] = v;
    }
  }
  __syncthreads();

  const int ly  = wv >> 1;
  const int lx  = (wv & 1) << 4;
  const int yy  = ly + 1;
  const int y   = y0 + ly;
  const int xxm = lx + 1 + nn;          // this lane's A-row pixel, LDS x
  const int xm  = x0 + lx + nn;         // global x

  // deg in {2,3,4}; exact reciprocals via selects (matches f32 divide)
  const int edges = (xm == 0) + (xm == 255) + (y == 0) + (y == 255);
  const float dinv = (edges == 0) ? 0.25f : ((edges == 1) ? (1.0f / 3.0f) : 0.5f);

  const float biasH = (nn < DD) ? (bh[nn] + bmp[nn]) : 0.f;
  const float br1n  = (nn < DD) ? br1[nn] : 0.f;
  const float w112  = (nn < DD) ? Wr1[nn * 13 + 12] : 0.f;
  const float br2n  = (nn < 3) ? br2[nn] : 0.f;
  const float ba0   = ba[0];

  // A-operand base: half-1 lanes are +2 channels = +408 floats
  const int baseH = yy * 34 + xxm + half * 408;

  // ---- GEMM1: conv3x3 (K=0..107) + deg-scaled neighbor sum (K=108..119) ----
  v8f acc = {0.f, 0.f, 0.f, 0.f, 0.f, 0.f, 0.f, 0.f};
  #pragma unroll
  for (int q = 0; q < 27; ++q) {
    const int kA = 4 * q;                       // half-0 k (even, c<=10)
    const int tA = kA / 12, cA = kA - 12 * tA;  // compile-time
    const int offA = cA * 204 + (tA / 3 - 1) * 34 + (tA - 3 * (tA / 3)) - 1;
    v2f a;
    a.x = sHid[baseH + offA];          // k   (channel cA [+2*half])
    a.y = sHid[baseH + offA + 204];    // k+1 (channel cA+1, same tap)
    v2f bb;
    bb.x = bpack[q * 64 + lane];
    bb.y = bpack[q * 64 + 32 + lane];
    acc = __builtin_amdgcn_wmma_f32_16x16x4_f32(false, a, false, bb,
                                                (short)0, acc, false, false);
  }
  #pragma unroll
  for (int q = 27; q < 30; ++q) {
    const int c0 = (q - 27) * 4;                // compile-time channel base
    const int o0 = c0 * 204;                    // + baseH gives (c0, yy, xxm)
    v2f a;
    a.x = dinv * (sHid[baseH + o0 - 34] + sHid[baseH + o0 + 34] +
                  sHid[baseH + o0 - 1]  + sHid[baseH + o0 + 1]);
    a.y = dinv * (sHid[baseH + o0 + 204 - 34] + sHid[baseH + o0 + 204 + 34] +
                  sHid[baseH + o0 + 204 - 1]  + sHid[baseH + o0 + 204 + 1]);
    v2f bb;
    bb.x = bpack[q * 64 + lane];
    bb.y = bpack[q * 64 + 32 + lane];
    acc = __builtin_amdgcn_wmma_f32_16x16x4_f32(false, a, false, bb,
                                                (short)0, acc, false, false);
  }

  // ---- residual + biases + ELU; stage hidden_new to per-wave T ----
  // D layout: vgpr j <-> pixel m=j+8*half, column nn
  float hnew[8];
  float* T = &sT[wv * 256];
  const int baseR = nn * 204 + yy * 34 + lx + 1 + 8 * half; // residual reads
  const int baseW = (8 * half) * 16 + nn;                   // T writes
  #pragma unroll
  for (int j = 0; j < 8; ++j) {
    float hv = (nn < DD) ? sHid[baseR + j] : 0.f;
    float s  = acc[j] + hv + biasH;
    float e  = (s > 0.f) ? s : (__expf(s) - 1.0f);   // ELU
    hnew[j]  = (nn < DD) ? e : 0.f;
    T[baseW + j * 16] = hnew[j];
  }

  // ---- GEMM2: [Wr1[:, :12] | Wa] on hidden_new (K=12) ----
  const int baseT = nn * 16 + 2 * half;   // A reads from T
  v8f acc2 = {0.f, 0.f, 0.f, 0.f, 0.f, 0.f, 0.f, 0.f};
  #pragma unroll
  for (int q = 0; q < 3; ++q) {
    v2f a;
    a.x = T[baseT + 4 * q];
    a.y = T[baseT + 4 * q + 1];
    v2f bb;
    bb.x = bpack[(30 + q) * 64 + lane];
    bb.y = bpack[(30 + q) * 64 + 32 + lane];
    acc2 = __builtin_amdgcn_wmma_f32_16x16x4_f32(false, a, false, bb,
                                                 (short)0, acc2, false, false);
  }

  float alivej[8], h1j[8];
  #pragma unroll
  for (int j = 0; j < 8; ++j) {
    float araw = __shfl(acc2[j], 12 + 16 * half, 32);      // alive pre-act (col 12)
    float al = 1.0f / (1.0f + __expf(-(araw + ba0)));
    alivej[j] = al;
    float v = acc2[j] + br1n + w112 * al;                  // Wr1 @ [h; alive] + br1
    h1j[j] = (nn < DD) ? fmaxf(v, 0.f) : 0.f;              // ReLU
  }
  #pragma unroll
  for (int j = 0; j < 8; ++j) {
    T[baseW + j * 16] = h1j[j];
  }

  // ---- GEMM3: Wr2 on h1 (K=12, 3 valid columns) ----
  v8f acc3 = {0.f, 0.f, 0.f, 0.f, 0.f, 0.f, 0.f, 0.f};
  #pragma unroll
  for (int q = 0; q < 3; ++q) {
    v2f a;
    a.x = T[baseT + 4 * q];
    a.y = T[baseT + 4 * q + 1];
    v2f bb;
    bb.x = bpack[(33 + q) * 64 + lane];
    bb.y = bpack[(33 + q) * 64 + 32 + lane];
    acc3 = __builtin_amdgcn_wmma_f32_16x16x4_f32(false, a, false, bb,
                                                 (short)0, acc3, false, false);
  }

  // ---- outputs staged to T = O[c][m], then coalesced global write ----
  #pragma unroll
  for (int j = 0; j < 8; ++j) {
    int mj = j + 8 * half;
    if (outFull) {
      float rgb = (nn < 3)
          ? alivej[j] * (1.0f / (1.0f + __expf(-(acc3[j] + br2n)))) : 0.f;
      if (nn == 0) T[0 * 16 + mj] = alivej[j];
      if (nn < 3)  T[(1 + nn) * 16 + mj] = rgb;
      if (nn < DD) T[(4 + nn) * 16 + mj] = hnew[j];
    } else {
      if (nn < DD) T[nn * 16 + mj] = hnew[j];
    }
  }

  const int outC = outFull ? 16 : DD;
  const size_t ob = (size_t)b * outC * NPIX;
  const int rowbase = y * 256 + x0 + lx;
  for (int i = lane; i < outC * 16; i += 32) {
    int c = i >> 4, mm = i & 15;
    out[ob + (size_t)c * NPIX + rowbase + mm] = T[c * 16 + mm];
  }
}

// ---------------------------------------------------------------------------
// Host launcher. Workspace: bpack (4096 floats) + hidden ping buffer
// (8*12*65536 floats ~ 25.2 MB). d_out doubles as the pong hidden buffer
// on odd steps; step 7 writes the full 16-channel state to d_out.
// Input dict order: x(0), edge_index(1), steps(2), Wh(3), bh(4), Wmp(5),
// bmp(6), Wa(7), ba(8), Wr1(9), br1(10), Wr2(11), br2(12).
// ---------------------------------------------------------------------------
extern "C" void kernel_launch(void* const* d_in, const int* in_sizes, int n_in,
                              void* d_out, int out_size, void* d_ws, size_t ws_size,
                              hipStream_t stream) {
  (void)in_sizes; (void)n_in; (void)out_size; (void)ws_size;
  const float* x   = (const float*)d_in[0];
  const float* Wh  = (const float*)d_in[3];
  const float* bh  = (const float*)d_in[4];
  const float* Wmp = (const float*)d_in[5];
  const float* bmp = (const float*)d_in[6];
  const float* Wa  = (const float*)d_in[7];
  const float* ba  = (const float*)d_in[8];
  const float* Wr1 = (const float*)d_in[9];
  const float* br1 = (const float*)d_in[10];
  const float* Wr2 = (const float*)d_in[11];
  const float* br2 = (const float*)d_in[12];
  float* outp = (float*)d_out;
  float* ws   = (float*)d_ws;

  float* bpack = ws;
  float* hbuf  = ws + 4096;

  build_bpack<<<1, 256, 0, stream>>>(Wh, Wmp, Wr1, Wa, Wr2, bpack);

  dim3 grid(256 / 32, 256 / 4, 8);
  dim3 blk(256);
  for (int k = 0; k < 8; ++k) {
    const float* inp; int inC, inOff;
    if (k == 0)      { inp = x;    inC = 16; inOff = 4; }
    else if (k & 1)  { inp = hbuf; inC = 12; inOff = 0; }
    else             { inp = outp; inC = 12; inOff = 0; }
    float* op; int full;
    if (k == 7)      { op = outp; full = 1; }
    else if (k & 1)  { op = outp; full = 0; }
    else             { op = hbuf; full = 0; }
    nca_step<<<grid, blk, 0, stream>>>(inp, inC, inOff, op, full, bpack,
                                       bh, bmp, Wr1, ba, br1, br2);
  }
}